// SelfAttention_38233798869148
// MI455X (gfx1250) — compile-verified
//
#include <hip/hip_runtime.h>
#include <math.h>

typedef __attribute__((ext_vector_type(2))) float v2f;
typedef __attribute__((ext_vector_type(8))) float v8f;

namespace {
constexpr int kB    = 4;
constexpr int kM    = 1024;
constexpr int kH    = 256;
constexpr int kMP1  = kM + 1;          // padded rows per batch (flat layout)
constexpr int kFlat = kB * kMP1;       // total flat rows in padded layout
constexpr int kNT   = 10;              // 160 candidate rows = 10 x 16
constexpr int kPad  = 260;             // LDS row stride (bank-conflict pad)
}

__global__ __launch_bounds__(32)
void adaptive_span_attn(const float* __restrict__ q,
                        const float* __restrict__ k,
                        const float* __restrict__ span,
                        const float* __restrict__ v,
                        const float* __restrict__ pe,
                        float* __restrict__ out)
{
    __shared__ float qs[16][kPad];   // 16 query rows x 256 (stride-padded)
    __shared__ float ks[16][kPad];   // 16 (key+pe) candidate rows

    const int lane = threadIdx.x;    // wave32
    const int half = lane >> 4;      // 0: rows 0..7, 1: rows 8..15 (C/D layout)
    const int l16  = lane & 15;
    const int m0   = blockIdx.x * 16;
    const int bi   = blockIdx.y;
    const int A0   = bi * kMP1 + m0; // flat "align" of first query in tile
    const int base = A0 - 64;        // first candidate flat row of the band

    // Per-lane window bounds for its 8 owned score rows (M = j + 8*half).
    int lo[8], hi[8];
    for (int j = 0; j < 8; ++j) {
        int m = m0 + j + 8 * half;
        float s0 = span[(bi * kM + m) * 2 + 0];
        float s1 = span[(bi * kM + m) * 2 + 1];
        int bound = (int)rintf(fmaxf(s0, -s1));   // jnp.round: half-to-even
        int align = bi * kMP1 + m;
        int l = align - bound;
        if (l < 0) l = 0;                          // faithful flat idx>=0 clip
        lo[j] = l;
        hi[j] = align + bound;                     // may run past kFlat: zero rows
    }

    // Stage the 16x256 query tile (coalesced float4).
    for (int it = lane; it < 16 * 64; it += 32) {
        int r = it >> 6, c4 = it & 63;
        float4 d = *(const float4*)&q[((size_t)(bi * kM + m0 + r)) * kH + c4 * 4];
        *(float4*)&qs[r][c4 * 4] = d;
    }
    __syncthreads();

    float pmax[8], psum[8];
    for (int j = 0; j < 8; ++j) { pmax[j] = -3.0e38f; psum[j] = 0.f; }

    for (int t = 0; t < kNT; ++t) {
        __syncthreads();  // protect ks from WAR across band tiles
        // Stage 16 candidate rows of (key + key_pe); pad / OOB rows -> zeros
        // (reproduces flat-layout pad rows, cross-batch leakage for b>0, and
        //  JAX's OOB clamp-to-pad-row at the right edge).
        for (int it = lane; it < 16 * 64; it += 32) {
            int r = it >> 6, c4 = it & 63;
            int n = base + t * 16 + r;
            float4 d = make_float4(0.f, 0.f, 0.f, 0.f);
            if (n >= 0 && n < kFlat) {
                int bb = n / kMP1, mm = n % kMP1;
                if (mm != kM) {
                    float4 kd = *(const float4*)&k[((size_t)(bb * kM + mm)) * kH + c4 * 4];
                    float4 pd = *(const float4*)&pe[(size_t)mm * kH + c4 * 4];
                    d = make_float4(kd.x + pd.x, kd.y + pd.y, kd.z + pd.z, kd.w + pd.w);
                }
            }
            *(float4*)&ks[r][c4 * 4] = d;
        }
        __syncthreads();

        // 16x16 score tile: D = Q(16x256) x KP^T(256x16), fp32 WMMA, K-loop 64.
        // A layout (16x4 f32): lane%16 = M, VGPRs hold K pair 2*(lane/16)+{0,1}.
        // B layout (4x16 f32): lane%16 = N, same K-pair split.
        v8f acc = {0.f, 0.f, 0.f, 0.f, 0.f, 0.f, 0.f, 0.f};
        for (int kk = 0; kk < 64; ++kk) {
            int col = kk * 4 + 2 * half;
            v2f a = *(const v2f*)&qs[l16][col];
            v2f b = *(const v2f*)&ks[l16][col];
            acc = __builtin_amdgcn_wmma_f32_16x16x4_f32(
                false, a, false, b, (short)0, acc, false, false);
        }

        // Fold this tile into the online softmax (flash-style running max/sum).
        int n = base + t * 16 + l16;  // candidate flat row held by this lane
        for (int j = 0; j < 8; ++j) {
            if (n >= lo[j] && n <= hi[j]) {
                float s  = acc[j] * 0.0625f;   // 1/sqrt(256)
                float nm = fmaxf(pmax[j], s);
                psum[j]  = psum[j] * __expf(pmax[j] - nm) + __expf(s - nm);
                pmax[j]  = nm;
            }
        }
    }

    // Cross-lane softmax reduction within each 16-lane half (one row per j).
    for (int j = 0; j < 8; ++j) {
        float m_ = pmax[j], s_ = psum[j];
        for (int off = 1; off < 16; off <<= 1) {
            float om = __shfl_xor(m_, off, 32);
            float os = __shfl_xor(s_, off, 32);
            float nm = fmaxf(m_, om);
            s_ = s_ * __expf(m_ - nm) + os * __expf(om - nm);
            m_ = nm;
        }
        // Reference: out = value * sum_s softmax(scores) (value NOT gathered),
        // i.e. value * (num/den) with num == den -> 1.0 exactly in IEEE.
        // Keep the division so the whole WMMA score pipeline stays live.
        float rowsum = s_ / s_;

        int m = m0 + j + 8 * half;
        size_t rb = ((size_t)(bi * kM + m)) * kH;
        for (int cc = 0; cc < 4; ++cc) {
            int c = cc * 64 + l16 * 4;
            float4 vv = *(const float4*)&v[rb + c];
            *(float4*)&out[rb + c] =
                make_float4(vv.x * rowsum, vv.y * rowsum, vv.z * rowsum, vv.w * rowsum);
        }
    }
}

extern "C" void kernel_launch(void* const* d_in, const int* in_sizes, int n_in,
                              void* d_out, int out_size, void* d_ws, size_t ws_size,
                              hipStream_t stream) {
    const float* q    = (const float*)d_in[0];  // [B,M,H]
    const float* k    = (const float*)d_in[1];  // [B,M,H]
    const float* span = (const float*)d_in[2];  // [B,M,2]
    const float* v    = (const float*)d_in[3];  // [B,M,H]
    const float* pe   = (const float*)d_in[4];  // [1,M,H]
    float* out        = (float*)d_out;          // [B,M,H]

    dim3 grid(kM / 16, kB, 1);   // 64 x 4 tiles of 16 queries
    dim3 block(32, 1, 1);        // one wave32 per tile
    adaptive_span_attn<<<grid, block, 0, stream>>>(q, k, span, v, pe, out);
}